// ProjMultiheadBlock_67551245631621
// MI455X (gfx1250) — compile-verified
//
#include <hip/hip_runtime.h>
#include <hip/hip_bf16.h>
#include <math.h>

// ---------------- problem constants ----------------
#define NB   32          // batch
#define NC   512         // channels
#define HWD  9409        // 97*97
#define HWP  9424        // HW padded to multiple of 16
#define NM   8           // mask heads
#define PL   32          // planes
#define LN_EPS 1e-5f

#define KCH  592         // split-K chunk (148 k-steps of 4)
#define NCHK 16          // ceil(9409/592) == ceil(9424/592) == 16
#define N16  (NB * NC * 16)   // elements of proj / ctx partial slab

typedef float v2f __attribute__((ext_vector_type(2)));
typedef float v8f __attribute__((ext_vector_type(8)));

static __device__ __forceinline__ v8f wmma_f32_4(v2f a, v2f b, v8f c) {
  // v_wmma_f32_16x16x4_f32 : D(16x16) = A(16x4) * B(4x16) + C
  return __builtin_amdgcn_wmma_f32_16x16x4_f32(false, a, false, b, (short)0, c,
                                               false, false);
}

// ---------------------------------------------------------------------------
// Kernel 1: proj partials. projP[chunk][b][c][m16] = sum_{hw in chunk} x*w_mask
// Per wave: two 16-channel tiles (share the w_mask B fragment), split-K.
// grid = NB * NCHK * 4 blocks of 128 (4 waves = 4 channel-pairs).
// ---------------------------------------------------------------------------
__global__ __launch_bounds__(128) void proj_kernel(
    const float* __restrict__ x, const float* __restrict__ w_mask,
    float* __restrict__ projP) {
  const int tid  = threadIdx.x;
  const int lane = tid & 31, wv = tid >> 5;
  const int b     = blockIdx.x >> 6;          // 64 blocks per batch
  const int r     = blockIdx.x & 63;
  const int chunk = r >> 2;
  const int cpair = ((r & 3) << 2) + wv;      // 0..15
  const int c0    = cpair << 5;               // 32 channels per wave
  const int q = lane & 15, half = lane >> 4;
  const float* arow0 = x + ((size_t)b * NC + c0 + q) * HWD + half * 2;
  const float* arow1 = arow0 + (size_t)16 * HWD;
  const float* brow  = w_mask + (size_t)(q & 7) * HWD + half * 2;
  // NOTE: no q<8 masking of B: garbage proj columns m>=8 only feed score rows
  // m>=8, which softmax overwrites with zeros.
  const int k_lo = chunk * KCH;
  const int k_hi = (k_lo + KCH < HWD) ? (k_lo + KCH) : HWD;
  const int k_full = k_hi - ((k_hi - k_lo) & 3);
  v8f acc0 = {}, acc1 = {};
  int k0 = k_lo;
  if (k0 + 4 <= k_full) {                     // software-pipelined main loop
    v2f a0c, a1c, bc;
    a0c.x = arow0[k0]; a0c.y = arow0[k0 + 1];
    a1c.x = arow1[k0]; a1c.y = arow1[k0 + 1];
    bc.x  = brow[k0];  bc.y  = brow[k0 + 1];
    for (; k0 + 8 <= k_full; k0 += 4) {
      const int kn = k0 + 4;
      v2f a0n, a1n, bn;
      a0n.x = arow0[kn]; a0n.y = arow0[kn + 1];
      a1n.x = arow1[kn]; a1n.y = arow1[kn + 1];
      bn.x  = brow[kn];  bn.y  = brow[kn + 1];
      acc0 = wmma_f32_4(a0c, bc, acc0);
      acc1 = wmma_f32_4(a1c, bc, acc1);
      a0c = a0n; a1c = a1n; bc = bn;
    }
    acc0 = wmma_f32_4(a0c, bc, acc0);
    acc1 = wmma_f32_4(a1c, bc, acc1);
    k0 += 4;
  }
  if (k0 < k_hi) {                            // ragged tail (last chunk only)
    const int k = k0 + half * 2;
    v2f a0, a1, bm;
    a0.x = (k     < k_hi) ? arow0[k0]     : 0.f;
    a0.y = (k + 1 < k_hi) ? arow0[k0 + 1] : 0.f;
    a1.x = (k     < k_hi) ? arow1[k0]     : 0.f;
    a1.y = (k + 1 < k_hi) ? arow1[k0 + 1] : 0.f;
    bm.x = (k     < k_hi) ? brow[k0]      : 0.f;
    bm.y = (k + 1 < k_hi) ? brow[k0 + 1]  : 0.f;
    acc0 = wmma_f32_4(a0, bm, acc0);
    acc1 = wmma_f32_4(a1, bm, acc1);
  }
  float* dst = projP + (size_t)chunk * N16;
#pragma unroll
  for (int g = 0; g < 8; ++g) {               // D row = g + half*8, col = q
    const int ch = c0 + g + half * 8;
    dst[((size_t)b * NC + ch) * 16 + q]        = acc0[g];
    dst[((size_t)b * NC + ch + 16) * 16 + q]   = acc1[g];
  }
}

// ---------------------------------------------------------------------------
// Kernel 1b: deterministic split-K reduction + mask bias.
// ---------------------------------------------------------------------------
__global__ __launch_bounds__(256) void proj_reduce_kernel(
    const float* __restrict__ projP, const float* __restrict__ b_mask,
    float* __restrict__ proj) {
  const unsigned stride = gridDim.x * blockDim.x;
  for (unsigned i = blockIdx.x * blockDim.x + threadIdx.x; i < (unsigned)N16;
       i += stride) {
    float s = 0.f;
#pragma unroll
    for (int c = 0; c < NCHK; ++c) s += projP[(size_t)c * N16 + i];
    const int n = i & 15;
    proj[i] = s + ((n < NM) ? b_mask[n] : 0.f);
  }
}

// ---------------------------------------------------------------------------
// Kernel 2: scores[b][m][hw] = sum_c proj[b][c][m] * x[b][c][hw]
// proj staged TRANSPOSED in LDS (padded rows, conflict-free, v2f ds loads).
// Per wave: two 16-hw tiles sharing the A fragment. K = C = 512.
// ---------------------------------------------------------------------------
#define SPJ_STRIDE 516
__global__ __launch_bounds__(256) void scores_kernel(
    const float* __restrict__ x, const float* __restrict__ proj,
    float* __restrict__ scores) {
  __shared__ float sp[16 * SPJ_STRIDE];       // ~33 KB
  const int tid = threadIdx.x;
  const int b   = blockIdx.x / 37;
  const int blk = blockIdx.x % 37;
  for (int i = tid; i < NC * 16; i += 256) {
    const int c = i >> 4, m = i & 15;
    sp[m * SPJ_STRIDE + c] = proj[(size_t)b * NC * 16 + i];
  }
  __syncthreads();
  const int lane = tid & 31, wv = tid >> 5;
  const int hw0  = blk * 256 + (wv << 5);     // two tiles: hw0, hw0+16
  if (hw0 >= HWP) return;
  const int q = lane & 15, half = lane >> 4;
  const int hwi0 = hw0 + q, hwi1 = hwi0 + 16;
  const float* xb = x + (size_t)b * NC * HWD + (size_t)(half * 2) * HWD;
  const float* ap = sp + q * SPJ_STRIDE + half * 2;   // A(m=q, k=c+half*2+j)
  v8f acc0 = {}, acc1 = {};
  if (hw0 + 31 < HWD) {                       // wave-uniform fast path
    v2f ac = *(const v2f*)&ap[0];
    v2f b0c, b1c;
    b0c.x = xb[hwi0]; b0c.y = xb[HWD + hwi0];
    b1c.x = xb[hwi1]; b1c.y = xb[HWD + hwi1];
    int c = 0;
    for (; c + 8 <= NC; c += 4) {
      const int cn = c + 4;
      v2f an = *(const v2f*)&ap[cn];
      const float* rn = xb + (size_t)cn * HWD;
      v2f b0n, b1n;
      b0n.x = rn[hwi0]; b0n.y = rn[HWD + hwi0];
      b1n.x = rn[hwi1]; b1n.y = rn[HWD + hwi1];
      acc0 = wmma_f32_4(ac, b0c, acc0);
      acc1 = wmma_f32_4(ac, b1c, acc1);
      ac = an; b0c = b0n; b1c = b1n;
    }
    acc0 = wmma_f32_4(ac, b0c, acc0);
    acc1 = wmma_f32_4(ac, b1c, acc1);
  } else {                                    // last tile pair, guarded
    const bool v0 = (hwi0 < HWD), v1 = (hwi1 < HWD);
    const int s0 = v0 ? hwi0 : 0, s1 = v1 ? hwi1 : 0;
    for (int c = 0; c < NC; c += 4) {
      v2f a = *(const v2f*)&ap[c];
      const float* rr = xb + (size_t)c * HWD;
      v2f b0, b1;
      b0.x = v0 ? rr[s0]       : 0.f;
      b0.y = v0 ? rr[HWD + s0] : 0.f;
      b1.x = v1 ? rr[s1]       : 0.f;
      b1.y = v1 ? rr[HWD + s1] : 0.f;
      acc0 = wmma_f32_4(a, b0, acc0);
      acc1 = wmma_f32_4(a, b1, acc1);
    }
  }
#pragma unroll
  for (int g = 0; g < 8; ++g) {
    const int m = g + half * 8;
    float* row = scores + ((size_t)b * 16 + m) * HWP;
    row[hw0 + q] = acc0[g];
    if (hw0 + 16 < HWP) row[hw0 + 16 + q] = acc1[g];
  }
}

// ---------------------------------------------------------------------------
// Kernel 3: softmax over HW per (b, m); zero padding + unused mask rows.
// ---------------------------------------------------------------------------
__global__ __launch_bounds__(256) void softmax_kernel(float* __restrict__ sc) {
  const int tid = threadIdx.x;
  const int m = blockIdx.x & 15, b = blockIdx.x >> 4;
  float* p = sc + ((size_t)b * 16 + m) * HWP;
  if (m >= NM) {
    for (int j = tid; j < HWP; j += 256) p[j] = 0.f;
    return;
  }
  __shared__ float red[256];
  float mx = -INFINITY;
  for (int j = tid; j < HWD; j += 256) mx = fmaxf(mx, p[j]);
  red[tid] = mx; __syncthreads();
  for (int s = 128; s > 0; s >>= 1) {
    if (tid < s) red[tid] = fmaxf(red[tid], red[tid + s]);
    __syncthreads();
  }
  mx = red[0]; __syncthreads();
  float sm = 0.f;
  for (int j = tid; j < HWD; j += 256) sm += __expf(p[j] - mx);
  red[tid] = sm; __syncthreads();
  for (int s = 128; s > 0; s >>= 1) {
    if (tid < s) red[tid] += red[tid + s];
    __syncthreads();
  }
  const float inv = 1.f / red[0];
  for (int j = tid; j < HWP; j += 256)
    p[j] = (j < HWD) ? __expf(p[j] - mx) * inv : 0.f;
}

// ---------------------------------------------------------------------------
// Kernel 4: context partials. ctxP[chunk][b][m][c] = sum_{hw in chunk} attn*x
// Per wave: two 16-channel tiles sharing the attn A fragment, split-K.
// grid = NB * NCHK * 4 blocks of 128.
// ---------------------------------------------------------------------------
__global__ __launch_bounds__(128) void context_kernel(
    const float* __restrict__ x, const float* __restrict__ attn,
    float* __restrict__ ctxP) {
  const int tid  = threadIdx.x;
  const int lane = tid & 31, wv = tid >> 5;
  const int b     = blockIdx.x >> 6;
  const int r     = blockIdx.x & 63;
  const int chunk = r >> 2;
  const int cpair = ((r & 3) << 2) + wv;
  const int c0    = cpair << 5;
  const int q = lane & 15, half = lane >> 4;
  const float* ar  = attn + ((size_t)b * 16 + q) * HWP + half * 2;
  const float* xr0 = x + ((size_t)b * NC + c0 + q) * HWD + half * 2;
  const float* xr1 = xr0 + (size_t)16 * HWD;
  const int h_lo = chunk * KCH;
  const int h_hi = (h_lo + KCH < HWP) ? (h_lo + KCH) : HWP;
  const int h_full = (h_hi < 9408) ? h_hi : 9408;   // x loads unguarded below
  v8f acc0 = {}, acc1 = {};
  int h0 = h_lo;
  if (h0 + 4 <= h_full) {                     // software-pipelined main loop
    v2f ac, b0c, b1c;
    ac.x  = ar[h0];  ac.y  = ar[h0 + 1];
    b0c.x = xr0[h0]; b0c.y = xr0[h0 + 1];
    b1c.x = xr1[h0]; b1c.y = xr1[h0 + 1];
    for (; h0 + 8 <= h_full; h0 += 4) {
      const int hn = h0 + 4;
      v2f an, b0n, b1n;
      an.x  = ar[hn];  an.y  = ar[hn + 1];
      b0n.x = xr0[hn]; b0n.y = xr0[hn + 1];
      b1n.x = xr1[hn]; b1n.y = xr1[hn + 1];
      acc0 = wmma_f32_4(ac, b0c, acc0);
      acc1 = wmma_f32_4(ac, b1c, acc1);
      ac = an; b0c = b0n; b1c = b1n;
    }
    acc0 = wmma_f32_4(ac, b0c, acc0);
    acc1 = wmma_f32_4(ac, b1c, acc1);
    h0 += 4;
  }
  for (; h0 < h_hi; h0 += 4) {                // guarded x tail (attn padded)
    const int h = h0 + half * 2;
    v2f a, b0, b1;
    a.x  = ar[h0]; a.y = ar[h0 + 1];
    b0.x = (h     < HWD) ? xr0[h0]     : 0.f;
    b0.y = (h + 1 < HWD) ? xr0[h0 + 1] : 0.f;
    b1.x = (h     < HWD) ? xr1[h0]     : 0.f;
    b1.y = (h + 1 < HWD) ? xr1[h0 + 1] : 0.f;
    acc0 = wmma_f32_4(a, b0, acc0);
    acc1 = wmma_f32_4(a, b1, acc1);
  }
  float* dst = ctxP + (size_t)chunk * N16;
#pragma unroll
  for (int g = 0; g < 8; ++g) {
    const int m = g + half * 8;
    dst[((size_t)b * 16 + m) * NC + c0 + q]      = acc0[g];
    dst[((size_t)b * 16 + m) * NC + c0 + 16 + q] = acc1[g];
  }
}

// ---------------------------------------------------------------------------
// Kernel 4b: deterministic split-K reduction for context.
// ---------------------------------------------------------------------------
__global__ __launch_bounds__(256) void ctx_reduce_kernel(
    const float* __restrict__ ctxP, float* __restrict__ ctx) {
  const unsigned stride = gridDim.x * blockDim.x;
  for (unsigned i = blockIdx.x * blockDim.x + threadIdx.x; i < (unsigned)N16;
       i += stride) {
    float s = 0.f;
#pragma unroll
    for (int c = 0; c < NCHK; ++c) s += ctxP[(size_t)c * N16 + i];
    ctx[i] = s;
  }
}

// ---------------------------------------------------------------------------
// Kernel 5: tiny MLP: conv1 -> LayerNorm -> ReLU -> conv2 -> add[b][c]
// ---------------------------------------------------------------------------
__global__ __launch_bounds__(256) void mlp_kernel(
    const float* __restrict__ ctx, const float* __restrict__ w1,
    const float* __restrict__ b1, const float* __restrict__ gam,
    const float* __restrict__ bet, const float* __restrict__ w2,
    const float* __restrict__ b2, float* __restrict__ addv) {
  __shared__ float part[256];
  __shared__ float h[PL];
  const int tid = threadIdx.x;
  const int b = blockIdx.x;
  const int p = tid & 31, m = tid >> 5;       // 8 heads x 32 planes
  const float* cr = ctx + ((size_t)b * 16 + m) * NC;
  const float* wr = w1 + (size_t)p * (NM * NC) + m * NC;
  float s = 0.f;
  for (int c = 0; c < NC; ++c) s += cr[c] * wr[c];
  part[(m << 5) + p] = s;
  __syncthreads();
  if (tid < PL) {
    float v = b1[tid];
    for (int mm = 0; mm < NM; ++mm) v += part[(mm << 5) + tid];
    h[tid] = v;
  }
  __syncthreads();
  float hn = 0.f;
  if (tid < PL) {
    float mu = 0.f;
    for (int i = 0; i < PL; ++i) mu += h[i];
    mu *= (1.f / PL);
    float var = 0.f;
    for (int i = 0; i < PL; ++i) { float d = h[i] - mu; var += d * d; }
    var *= (1.f / PL);
    hn = (h[tid] - mu) * rsqrtf(var + LN_EPS) * gam[tid] + bet[tid];
    hn = fmaxf(hn, 0.f);
  }
  __syncthreads();
  if (tid < PL) h[tid] = hn;
  __syncthreads();
  for (int c = tid; c < NC; c += 256) {
    float a = b2[c];
    const float* w2r = w2 + (size_t)c * PL;
    for (int pp = 0; pp < PL; ++pp) a += h[pp] * w2r[pp];
    addv[(size_t)b * NC + c] = a;
  }
}

// ---------------------------------------------------------------------------
// Kernel 6: out = x + add[b][c] (broadcast over HW), float4 streaming.
// ---------------------------------------------------------------------------
__global__ __launch_bounds__(256) void bcast_add_kernel(
    const float* __restrict__ x, const float* __restrict__ addv,
    float* __restrict__ out) {
  const unsigned n4 = (unsigned)(((size_t)NB * NC * HWD) / 4);
  const unsigned stride = gridDim.x * blockDim.x;
  for (unsigned i = blockIdx.x * blockDim.x + threadIdx.x; i < n4; i += stride) {
    const float4 v = reinterpret_cast<const float4*>(x)[i];
    const unsigned e = i * 4u;
    float4 r;
    r.x = v.x + addv[(e    ) / HWD];   // constant division -> umulhi
    r.y = v.y + addv[(e + 1) / HWD];
    r.z = v.z + addv[(e + 2) / HWD];
    r.w = v.w + addv[(e + 3) / HWD];
    reinterpret_cast<float4*>(out)[i] = r;
  }
}

// ---------------- workspace layout (bytes) ----------------
static constexpr size_t SZ_PART = (size_t)NCHK * N16 * 4;     // 16 MB (reused)
static constexpr size_t SZ_PROJ = (size_t)N16 * 4;            // 1 MB
static constexpr size_t SZ_SC   = (size_t)NB * 16 * HWP * 4;  // ~18.4 MB
static constexpr size_t SZ_CTX  = (size_t)N16 * 4;            // 1 MB
static constexpr size_t OFF_PART = 0;
static constexpr size_t OFF_PROJ = OFF_PART + SZ_PART;
static constexpr size_t OFF_SC   = OFF_PROJ + SZ_PROJ;
static constexpr size_t OFF_CTX  = OFF_SC + SZ_SC;
static constexpr size_t OFF_ADD  = OFF_CTX + SZ_CTX;

extern "C" void kernel_launch(void* const* d_in, const int* in_sizes, int n_in,
                              void* d_out, int out_size, void* d_ws, size_t ws_size,
                              hipStream_t stream) {
  const float* x      = (const float*)d_in[0];
  const float* w_mask = (const float*)d_in[1];
  const float* b_mask = (const float*)d_in[2];
  const float* w1     = (const float*)d_in[3];
  const float* b1     = (const float*)d_in[4];
  const float* gam    = (const float*)d_in[5];
  const float* bet    = (const float*)d_in[6];
  const float* w2     = (const float*)d_in[7];
  const float* b2     = (const float*)d_in[8];
  float* out = (float*)d_out;
  char*  ws  = (char*)d_ws;
  float* partials = (float*)(ws + OFF_PART);   // projP, then reused as ctxP
  float* proj     = (float*)(ws + OFF_PROJ);
  float* attn     = (float*)(ws + OFF_SC);
  float* ctx      = (float*)(ws + OFF_CTX);
  float* addv     = (float*)(ws + OFF_ADD);

  proj_kernel       <<<NB * NCHK * 4, 128, 0, stream>>>(x, w_mask, partials);
  proj_reduce_kernel<<<256,           256, 0, stream>>>(partials, b_mask, proj);
  scores_kernel     <<<NB * 37,       256, 0, stream>>>(x, proj, attn);
  softmax_kernel    <<<NB * 16,       256, 0, stream>>>(attn);
  context_kernel    <<<NB * NCHK * 4, 128, 0, stream>>>(x, attn, partials);
  ctx_reduce_kernel <<<256,           256, 0, stream>>>(partials, ctx);
  mlp_kernel        <<<NB,            256, 0, stream>>>(ctx, w1, b1, gam, bet, w2, b2, addv);
  bcast_add_kernel  <<<2048,          256, 0, stream>>>(x, addv, out);
}